// MappedConvolution_34282428956679
// MI455X (gfx1250) — compile-verified
//
#include <hip/hip_runtime.h>

typedef float v2f __attribute__((ext_vector_type(2)));
typedef float v8f __attribute__((ext_vector_type(8)));

#define BB    2
#define CIN   8
#define COUT  16
#define HH    512
#define WW    1024
#define KTAP  9
#define KD    (CIN * KTAP)      // 72 = contraction depth
#define NPIX  (HH * WW)         // 524288 pixels per (b, plane)
#define TILE  16                // pixels per wave (WMMA M dim)
#define WAVES 4                 // waves per block

// Compiler-visible ordering fence for same-wave LDS produce->consume.
// DS ops are in-order per wave on CDNA5; this stops the compiler reordering.
__device__ __forceinline__ void lds_fence() {
    asm volatile("s_wait_dscnt 0" ::: "memory");
}

__global__ __launch_bounds__(WAVES * 32)
void mapped_conv_wmma(const float* __restrict__ x,      // [B, CIN, H, W]
                      const float* __restrict__ smap,   // [OH, OW, K, 2]
                      const float* __restrict__ wgt,    // [COUT, CIN, K]
                      const float* __restrict__ bias,   // [COUT]
                      float* __restrict__ out)          // [B, COUT, OH, OW]
{
    // S[b][kk/2][pixel][kk&1]: A-fragment pairs (kk, kk+1) contiguous -> ds b64
    __shared__ float S[WAVES][BB][KD / 2][TILE][2];  // 4*2*36*16*2*4 = 36864 B

    const int lane    = threadIdx.x & 31;
    const int wave    = threadIdx.x >> 5;
    const int pixbase = (blockIdx.x * WAVES + wave) * TILE;

    float (*Sw)[KD / 2][TILE][2] = S[wave];

    // ---------------- Stage A: bilinear gather into LDS ----------------
    // 16 pixels x 9 taps = 144 sample points over 32 lanes, 5 unrolled steps.
    // Last half-step clamps to point 143 (duplicate work, same value/address)
    // so control flow stays uniform and all smap loads can issue up front.
    #pragma unroll
    for (int it = 0; it < 5; ++it) {
        int s = it * 32 + lane;
        s = s > TILE * KTAP - 1 ? TILE * KTAP - 1 : s;
        const int p = (int)((unsigned)s / KTAP);
        const int k = s - p * KTAP;
        // coalesced: flat (px,py) pairs are contiguous in s
        const float* sm = smap + ((size_t)pixbase * KTAP + s) * 2;
        const float px = sm[0], py = sm[1];
        const float x0f = floorf(px), y0f = floorf(py);
        const float wx = px - x0f, wy = py - y0f;
        int ix = (int)x0f, iy = (int)y0f;
        // input coords are strictly < W-1 / H-1, so x0+1 / y0+1 in-bounds
        ix = ix < 0 ? 0 : (ix > WW - 2 ? WW - 2 : ix);
        iy = iy < 0 ? 0 : (iy > HH - 2 ? HH - 2 : iy);
        const float w00 = (1.f - wy) * (1.f - wx);
        const float w01 = (1.f - wy) * wx;
        const float w10 = wy * (1.f - wx);
        const float w11 = wy * wx;
        const float* base = x + (size_t)iy * WW + ix;
        // reuse coords + weights across all 16 (batch, channel) planes
        #pragma unroll
        for (int b = 0; b < BB; ++b) {
            #pragma unroll
            for (int c = 0; c < CIN; ++c) {
                const float* pl = base + (size_t)(b * CIN + c) * NPIX;
                const float t0 = pl[0],  t1 = pl[1];
                const float u0 = pl[WW], u1 = pl[WW + 1];
                const int kk = c * KTAP + k;
                Sw[b][kk >> 1][p][kk & 1] =
                    w00 * t0 + w01 * t1 + w10 * u0 + w11 * u1;
            }
        }
    }
    lds_fence();

    // ------------- Stage B: 16x16x72 fp32 GEMM via WMMA -------------
    // A (16x4 f32): lanes 0-15 -> M=lane, K={0,1}; lanes 16-31 -> M=lane-16, K={2,3}
    // B (4x16 f32): mirrored with N in place of M
    // C/D (16x16):  N = lane&15 (const per lane), M = vgpr + 8*(lane>>4)
    const int n  = lane & 15;
    const int hi = lane >> 4;
    const float bv = bias[n];           // N constant per lane -> bias broadcast
    v8f acc[BB];
    #pragma unroll
    for (int b = 0; b < BB; ++b)
        #pragma unroll
        for (int j = 0; j < 8; ++j) acc[b][j] = bv;

    #pragma unroll
    for (int t = 0; t < KD / 4; ++t) {  // 18 K-chunks of 4; kk = 4t + 2*hi (even)
        const int kh = t * 2 + hi;      // pair row in S
        const v2f bw = *(const v2f*)&wgt[n * KD + t * 4 + hi * 2];  // 8B aligned
        const v2f a0 = *(const v2f*)&Sw[0][kh][n][0];               // ds b64
        const v2f a1 = *(const v2f*)&Sw[1][kh][n][0];
        acc[0] = __builtin_amdgcn_wmma_f32_16x16x4_f32(
            false, a0, false, bw, (short)0, acc[0], false, false);
        acc[1] = __builtin_amdgcn_wmma_f32_16x16x4_f32(
            false, a1, false, bw, (short)0, acc[1], false, false);
    }

    // ------------- Stage C: LDS transpose for coalesced b64 stores -------------
    float (*D)[TILE] = (float (*)[TILE]) & Sw[0][0][0];  // 16x16 staging (reuse)
    const int o0 = lane >> 3;            // 0..3
    const int pp = (lane & 7) * 2;       // pixel-pair base
    #pragma unroll
    for (int b = 0; b < BB; ++b) {
        #pragma unroll
        for (int j = 0; j < 8; ++j)
            D[n][hi * 8 + j] = acc[b][j];   // D[out_ch][pixel]
        lds_fence();
        #pragma unroll
        for (int r = 0; r < 4; ++r) {       // 4 channel-rows per store instr
            const int o = r * 4 + o0;
            const v2f dv = *(const v2f*)&D[o][pp];
            *(v2f*)&out[((size_t)(b * COUT + o)) * NPIX + pixbase + pp] = dv;
        }
        lds_fence();
    }
}

extern "C" void kernel_launch(void* const* d_in, const int* in_sizes, int n_in,
                              void* d_out, int out_size, void* d_ws, size_t ws_size,
                              hipStream_t stream) {
    (void)in_sizes; (void)n_in; (void)out_size; (void)d_ws; (void)ws_size;
    const float* x    = (const float*)d_in[0];
    const float* smap = (const float*)d_in[1];
    const float* wgt  = (const float*)d_in[2];
    const float* bias = (const float*)d_in[3];
    float* out        = (float*)d_out;

    const int tiles  = NPIX / TILE;       // 32768 16-pixel tiles
    const int blocks = tiles / WAVES;     // 8192 blocks of 4 waves
    mapped_conv_wmma<<<blocks, WAVES * 32, 0, stream>>>(x, smap, wgt, bias, out);
}